// FilterLayer_80135499809183
// MI455X (gfx1250) — compile-verified
//
#include <hip/hip_runtime.h>

typedef __attribute__((ext_vector_type(2))) float v2f;
typedef __attribute__((ext_vector_type(8))) float v8f;

#define HTOT 512
#define WTOT 512
#define PADH 8
#define KK2  289
#define FDIM 100
#define ODIM 180

__device__ __forceinline__ unsigned encodeF(float f) {
    unsigned b = __float_as_uint(f);
    return (b & 0x80000000u) ? ~b : (b | 0x80000000u);
}
__device__ __forceinline__ float decodeF(unsigned e) {
    unsigned b = (e & 0x80000000u) ? (e ^ 0x80000000u) : ~e;
    return __uint_as_float(b);
}

__global__ void init_minmax(unsigned* ws) {
    ws[0] = 0xFFFFFFFFu;  // encoded-min init
    ws[1] = 0u;           // encoded-max init
}

// Block: 128 threads = 4 waves. Each wave computes 16 consecutive pixels of one
// image row via the 16x16x4 f32 WMMA diagonal-batch trick (exact f32 math).
// Inner loop uses incremental row-wrap addressing: no divisions, ~6 VALU ops
// per WMMA so the matrix pipe / LDS are the limiters, not VALU.
__global__ __launch_bounds__(128) void filter_wmma(
    const float* __restrict__ img,
    const int*   __restrict__ freq,
    const int*   __restrict__ orient,
    const float* __restrict__ fb,
    float*       __restrict__ yout,
    unsigned*    __restrict__ ws)
{
    __shared__ float tile[17 * 80];     // rows h-8..h+8, cols wblk-8..wblk+71
    __shared__ float diag_s[4][16];

    const int tid  = threadIdx.x;
    const int wave = tid >> 5;
    const int lane = tid & 31;
    const int h    = blockIdx.y;
    const int wblk = blockIdx.x * 64;

    // Cooperative zero-padded tile load (bounds handled here, not in hot loop).
    for (int i = tid; i < 17 * 80; i += 128) {
        int r = i / 80, c = i - 80 * (i / 80);
        int gr = h - PADH + r;
        int gc = wblk - PADH + c;
        float v = 0.0f;
        if (gr >= 0 && gr < HTOT && gc >= 0 && gc < WTOT)
            v = img[gr * WTOT + gc];
        tile[i] = v;
    }
    __syncthreads();

    const int p       = lane & 15;          // pixel within wave's group of 16
    const int off     = (lane >> 4) * 2;    // lanes 0-15 -> k+0,1 ; 16-31 -> k+2,3
    const int w       = wblk + wave * 16 + p;
    const int colbase = wave * 16 + p;      // tile col for s=0 of this pixel

    int fi = freq[h * WTOT + w] - 1;   fi = min(max(fi, 0), FDIM - 1);
    int oi = orient[h * WTOT + w] - 1; oi = min(max(oi, 0), ODIM - 1);
    const float* kbase = fb + (size_t)(fi * ODIM + oi) * KK2;

    // Incremental patch-walk state for element k0 = 4*c + off:
    //   s0 = k0 % 17, ap -> tile[r0*80 + colbase + s0]
    const float* ap   = &tile[colbase + off];
    const float* kptr = kbase + off;
    int s0 = off;

    v8f acc = {0.f, 0.f, 0.f, 0.f, 0.f, 0.f, 0.f, 0.f};
    #pragma unroll 8
    for (int c = 0; c < 72; ++c) {
        v2f a, b;
        a.x = ap[0];                               // patch_p[k0]
        const float* ayp = ap + ((s0 == 16) ? 64 : 1);  // k0+1 (row wrap: -16+80)
        a.y = *ayp;                                // patch_p[k0+1]
        b.x = kptr[0];                             // kernel_p[k0]
        b.y = kptr[1];                             // kernel_p[k0+1]
        acc = __builtin_amdgcn_wmma_f32_16x16x4_f32(
            /*neg_a=*/false, a, /*neg_b=*/false, b,
            /*c_mod=*/(short)0, acc, /*reuse_a=*/false, /*reuse_b=*/false);
        // advance 4 elements; at most one row wrap (s0 <= 20 before wrap)
        s0 += 4;
        int wrap = (s0 >= 17);
        s0 -= wrap ? 17 : 0;
        ap += wrap ? 67 : 4;                       // 4 + (80-17)
        kptr += 4;
    }

    // Extract diagonal D[p][p]: lane p holds it in VGPR p (p<8);
    // lane p+16 holds it in VGPR p-8 (p>=8).
    int sel = (lane < 8) ? lane : (lane - 24);
    float dv = acc[0];
    #pragma unroll
    for (int i = 1; i < 8; ++i) dv = (sel == i) ? acc[i] : dv;
    if (lane < 8)        diag_s[wave][lane]      = dv;
    else if (lane >= 24) diag_s[wave][lane - 16] = dv;
    __syncthreads();

    // Tail element k=288 (r=16, s=16), then store + min/max reduction.
    float y = diag_s[wave][p] + tile[16 * 80 + colbase + 16] * kbase[288];
    if (lane < 16) yout[h * WTOT + w] = y;

    float mn = y, mx = y;   // lanes 16-31 duplicate lanes 0-15: harmless
    #pragma unroll
    for (int d = 16; d >= 1; d >>= 1) {
        mn = fminf(mn, __shfl_xor(mn, d, 32));
        mx = fmaxf(mx, __shfl_xor(mx, d, 32));
    }
    if (lane == 0) {
        atomicMin(&ws[0], encodeF(mn));
        atomicMax(&ws[1], encodeF(mx));
    }
}

// In-place finalize: norm -> sigmoid -> norm. Second norm's extrema are the
// sigmoid images of the first norm's extrema (monotonicity), so one pass.
__global__ __launch_bounds__(256) void finalize(
    float* __restrict__ y, const unsigned* __restrict__ ws)
{
    int idx = blockIdx.x * 256 + threadIdx.x;
    float ymin  = decodeF(ws[0]);
    float ymax  = decodeF(ws[1]);
    float range = ymax - ymin;
    float denom = fmaxf(range, 1e-8f);
    float n1    = 100.0f * (y[idx] - ymin) / denom;
    float s     = 100.0f / (1.0f + expf(11.0f - 0.2f * n1));
    float n1max = 100.0f * range / denom;          // == 100 unless degenerate
    float smin  = 100.0f / (1.0f + expf(11.0f));   // s at n1 = 0
    float smax  = 100.0f / (1.0f + expf(11.0f - 0.2f * n1max));
    y[idx] = 100.0f * (s - smin) / fmaxf(smax - smin, 1e-8f);
}

extern "C" void kernel_launch(void* const* d_in, const int* in_sizes, int n_in,
                              void* d_out, int out_size, void* d_ws, size_t ws_size,
                              hipStream_t stream) {
    const float* img    = (const float*)d_in[0];
    const int*   freq   = (const int*)d_in[1];
    const int*   orient = (const int*)d_in[2];
    const float* fb     = (const float*)d_in[3];
    float*       out    = (float*)d_out;
    unsigned*    ws     = (unsigned*)d_ws;

    init_minmax<<<1, 1, 0, stream>>>(ws);

    dim3 grid1(WTOT / 64, HTOT);
    filter_wmma<<<grid1, 128, 0, stream>>>(img, freq, orient, fb, out, ws);

    finalize<<<(HTOT * WTOT) / 256, 256, 0, stream>>>(out, ws);
}